// PEPSLayer_82300163326807
// MI455X (gfx1250) — compile-verified
//
#include <hip/hip_runtime.h>

typedef float v2f __attribute__((ext_vector_type(2)));
typedef float v8f __attribute__((ext_vector_type(8)));

#define BLOCK   256
#define NWAVES  8
#define STATE   7776      // 6^5 state elements
#define MSZ     1296      // 36*36 site matrix
#define PM_FLOATS 62208   // 24 sites * 2 f * 1296  (rearranged peps)
#define CM_FLOATS 25920   // 2 f * 10 o * 1296      (rearranged center)
#define SM_FLOATS (3 * STATE + MSZ + BLOCK)

// ---------------------------------------------------------------------------
// Prep: rearrange peps[i,j,f,u,l,d,r] -> Pm[site,f, d*6+r, l*6+u]
//       rearrange center[f,u,l,d,r,o] -> Cm[f,o, d*6+r, l*6+u]
// ---------------------------------------------------------------------------
__global__ void __launch_bounds__(BLOCK)
peps_prep(const float* __restrict__ peps, const float* __restrict__ center,
          float* __restrict__ ws) {
  int idx = blockIdx.x * BLOCK + threadIdx.x;
  if (idx < PM_FLOATS) {
    int e  = idx;
    int lu = e % 36, l = lu / 6, u = lu % 6;
    int dr = (e / 36) % 36, d = dr / 6, r = dr % 6;
    int f  = (e / 1296) % 2;
    int s  = e / 2592;                       // site = i*4 + j
    ws[idx] = peps[((((s * 2 + f) * 6 + u) * 6 + l) * 6 + d) * 6 + r];
  } else if (idx < PM_FLOATS + CM_FLOATS) {
    int e  = idx - PM_FLOATS;
    int lu = e % 36, l = lu / 6, u = lu % 6;
    int dr = (e / 36) % 36, d = dr / 6, r = dr % 6;
    int o  = (e / 1296) % 10;
    int f  = e / 12960;
    ws[idx] = center[(((((f * 6 + u) * 6 + l) * 6 + d) * 6 + r) * 10) + o];
  }
}

// ---------------------------------------------------------------------------
// One site contraction via V_WMMA_F32_16X16X4_F32 (fp32 WMMA, K=36 = 9x4).
// Route A (post-major): out[p, dr, c] = sum_lu M[dr,lu] * S[p, lu, c]
// Route B (pre-major):  out[row, dr, q] = sum_lu S[row, lu, q] * M[dr,lu]
// ---------------------------------------------------------------------------
__device__ void site_contract(const float* __restrict__ src,
                              float* __restrict__ dst,
                              const float* __restrict__ Mlds,
                              int pre, int post, bool routeB,
                              int wave, int lane) {
  const int lid   = lane & 15;
  const int kbase = (lane >> 4) << 1;  // 0 or 2 (A/B K split across lane halves)
  const int hi8   = (lane >> 4) << 3;  // 0 or 8 (C/D row split)

  if (!routeB) {
    const int colTiles = (post + 15) >> 4;
    const int total = pre * 3 * colTiles;
    for (int task = wave; task < total; task += NWAVES) {
      const int p  = task / (3 * colTiles);
      const int rm = task % (3 * colTiles);
      const int rt = rm / colTiles;   // dr row tile (36 -> 3 tiles)
      const int ct = rm % colTiles;   // post col tile
      const int row = rt * 16 + lid;  // dr
      const int col = ct * 16 + lid;  // post column
      const float am = (row < 36)   ? 1.0f : 0.0f;
      const float bm = (col < post) ? 1.0f : 0.0f;
      const int rowC = (row < 36)   ? row : 0;
      const int colC = (col < post) ? col : 0;
      const float* sp = src + p * 36 * post;
      v8f acc = {};
#pragma unroll
      for (int k0 = 0; k0 < 36; k0 += 4) {
        const int k = k0 + kbase;
        v2f a, bb;
        a.x  = Mlds[rowC * 36 + k] * am;         // A = M rows (contiguous in lu)
        a.y  = Mlds[rowC * 36 + k + 1] * am;
        bb.x = sp[k * post + colC] * bm;         // B = state columns
        bb.y = sp[(k + 1) * post + colC] * bm;
        acc = __builtin_amdgcn_wmma_f32_16x16x4_f32(false, a, false, bb,
                                                    (short)0, acc, false, false);
      }
      if (col < post) {
        float* dp = dst + p * 36 * post + col;
        const int rbase = rt * 16 + hi8;
#pragma unroll
        for (int v = 0; v < 8; ++v) {
          const int crow = rbase + v;
          if (crow < 36) dp[crow * post] = acc[v];
        }
      }
    }
  } else {
    const int rowTiles = (pre + 15) >> 4;
    const int total = post * rowTiles * 3;
    for (int task = wave; task < total; task += NWAVES) {
      const int q  = task / (rowTiles * 3);
      const int rm = task % (rowTiles * 3);
      const int rt = rm / 3;          // pre row tile
      const int ct = rm % 3;          // dr col tile (36 -> 3 tiles)
      const int row = rt * 16 + lid;  // pre row
      const int col = ct * 16 + lid;  // dr column
      const float am = (row < pre) ? 1.0f : 0.0f;
      const float bm = (col < 36)  ? 1.0f : 0.0f;
      const int rowC = (row < pre) ? row : 0;
      const int colC = (col < 36)  ? col : 0;
      v8f acc = {};
#pragma unroll
      for (int k0 = 0; k0 < 36; k0 += 4) {
        const int k = k0 + kbase;
        v2f a, bb;
        a.x  = src[(rowC * 36 + k) * post + q] * am;     // A = state rows
        a.y  = src[(rowC * 36 + k + 1) * post + q] * am;
        bb.x = Mlds[colC * 36 + k] * bm;                 // B = M^T (contig in lu)
        bb.y = Mlds[colC * 36 + k + 1] * bm;
        acc = __builtin_amdgcn_wmma_f32_16x16x4_f32(false, a, false, bb,
                                                    (short)0, acc, false, false);
      }
      if (col < 36) {
        const int rbase = rt * 16 + hi8;
#pragma unroll
        for (int v = 0; v < 8; ++v) {
          const int crow = rbase + v;
          if (crow < pre) dst[(crow * 36 + col) * post + q] = acc[v];
        }
      }
    }
  }
}

// Row end: sum over last bond (axis 4), then prepend delta(0) as new index 0.
__device__ void row_end(const float* __restrict__ in, float* __restrict__ out,
                        int tid) {
  for (int t = tid; t < STATE; t += BLOCK) {
    float v = 0.0f;
    if (t < 1296) {
      const float* p = in + t * 6;
      v = ((p[0] + p[1]) + (p[2] + p[3])) + (p[4] + p[5]);
    }
    out[t] = v;
  }
}

__device__ void do_site(const float* __restrict__ M0, const float* __restrict__ M1,
                        float x0, float x1,
                        const float* __restrict__ src, float* __restrict__ dst,
                        float* __restrict__ Mlds,
                        int pre, int post, bool routeB,
                        int tid, int wave, int lane) {
  for (int e = tid; e < MSZ; e += BLOCK) Mlds[e] = x0 * M0[e] + x1 * M1[e];
  __syncthreads();
  site_contract(src, dst, Mlds, pre, post, routeB, wave, lane);
  __syncthreads();
}

// ---------------------------------------------------------------------------
// Main: one workgroup (8 wave32) per batch element.
// ---------------------------------------------------------------------------
__global__ void __launch_bounds__(BLOCK)
peps_main(const float* __restrict__ inputs, const float* __restrict__ ws,
          float* __restrict__ out) {
  extern __shared__ float smem[];
  float* Mlds = smem + 3 * STATE;
  float* red  = smem + 3 * STATE + MSZ;
  const int b    = blockIdx.x;
  const int tid  = threadIdx.x;
  const int wave = tid >> 5;
  const int lane = tid & 31;
  const float* Pm = ws;
  const float* Cm = ws + PM_FLOATS;
  const int P6[4] = {1, 6, 36, 216};

  // initial boundary state: delta at flat index 0
  for (int t = tid; t < STATE; t += BLOCK) smem[t] = (t == 0) ? 1.0f : 0.0f;
  __syncthreads();

  // ---- shared prefix: rows 0..2 fully, row 3 sites j=0,1 (ping-pong buf 0/1)
  int cur = 0;
  for (int i = 0; i < 4; ++i) {
    const int jmax = (i == 3) ? 2 : 4;
    for (int j = 0; j < jmax; ++j) {
      const int site = i * 4 + j;
      const float x0 = inputs[((b * 2 + 0) * 6 + i) * 4 + j];
      const float x1 = inputs[((b * 2 + 1) * 6 + i) * 4 + j];
      do_site(Pm + site * 2592, Pm + site * 2592 + MSZ, x0, x1,
              smem + cur * STATE, smem + (cur ^ 1) * STATE, Mlds,
              P6[j], P6[3 - j], j >= 2, tid, wave, lane);
      cur ^= 1;
    }
    if (i < 3) {
      row_end(smem + cur * STATE, smem + (cur ^ 1) * STATE, tid);
      __syncthreads();
      cur ^= 1;
    }
  }
  const int Pbuf = cur;        // prefix state, pinned for all o
  const int U = cur ^ 1;       // free buffer in {0,1}
  const int V = 2;             // free buffer

  const float xc0 = inputs[((b * 2 + 0) * 6 + 3) * 4 + 2];
  const float xc1 = inputs[((b * 2 + 1) * 6 + 3) * 4 + 2];

  // ---- per output-channel pass through center + remaining sites
  for (int o = 0; o < 10; ++o) {
    // center site (3,2): pre=36, post=6, route B
    do_site(Cm + o * MSZ, Cm + (10 + o) * MSZ, xc0, xc1,
            smem + Pbuf * STATE, smem + U * STATE, Mlds,
            36, 6, true, tid, wave, lane);
    int c2 = U, n2 = V, t;
    // site (3,3): pre=216, post=1, route B
    {
      const float x0 = inputs[((b * 2 + 0) * 6 + 3) * 4 + 3];
      const float x1 = inputs[((b * 2 + 1) * 6 + 3) * 4 + 3];
      do_site(Pm + 15 * 2592, Pm + 15 * 2592 + MSZ, x0, x1,
              smem + c2 * STATE, smem + n2 * STATE, Mlds,
              216, 1, true, tid, wave, lane);
      t = c2; c2 = n2; n2 = t;
    }
    row_end(smem + c2 * STATE, smem + n2 * STATE, tid);
    __syncthreads();
    t = c2; c2 = n2; n2 = t;

    for (int i = 4; i < 6; ++i) {
      for (int j = 0; j < 4; ++j) {
        const int site = i * 4 + j;
        const float x0 = inputs[((b * 2 + 0) * 6 + i) * 4 + j];
        const float x1 = inputs[((b * 2 + 1) * 6 + i) * 4 + j];
        do_site(Pm + site * 2592, Pm + site * 2592 + MSZ, x0, x1,
                smem + c2 * STATE, smem + n2 * STATE, Mlds,
                P6[j], P6[3 - j], j >= 2, tid, wave, lane);
        t = c2; c2 = n2; n2 = t;
      }
      if (i < 5) {
        row_end(smem + c2 * STATE, smem + n2 * STATE, tid);
        __syncthreads();
        t = c2; c2 = n2; n2 = t;
      }
    }

    // final full reduction of the state -> out[b, o]
    float s = 0.0f;
    for (int u = tid; u < STATE; u += BLOCK) s += smem[c2 * STATE + u];
    red[tid] = s;
    __syncthreads();
    for (int off = BLOCK / 2; off > 0; off >>= 1) {
      if (tid < off) red[tid] += red[tid + off];
      __syncthreads();
    }
    if (tid == 0) out[b * 10 + o] = red[0];
    __syncthreads();
  }
}

// ---------------------------------------------------------------------------
extern "C" void kernel_launch(void* const* d_in, const int* in_sizes, int n_in,
                              void* d_out, int out_size, void* d_ws, size_t ws_size,
                              hipStream_t stream) {
  const float* inputs = (const float*)d_in[0];  // (B=256, F=2, X=6, Y=4)
  const float* peps   = (const float*)d_in[1];  // (6,4,2,6,6,6,6)
  const float* center = (const float*)d_in[2];  // (2,6,6,6,6,10)
  float* outp = (float*)d_out;                  // (256, 10)
  float* ws   = (float*)d_ws;

  const int prepN = PM_FLOATS + CM_FLOATS;
  peps_prep<<<(prepN + BLOCK - 1) / BLOCK, BLOCK, 0, stream>>>(peps, center, ws);

  const size_t smemBytes = (size_t)SM_FLOATS * sizeof(float);  // ~97 KB
  (void)hipFuncSetAttribute((const void*)peps_main,
                            hipFuncAttributeMaxDynamicSharedMemorySize,
                            (int)smemBytes);
  peps_main<<<256, BLOCK, smemBytes, stream>>>(inputs, ws, outp);
}